// DM_28166395527920
// MI455X (gfx1250) — compile-verified
//
#include <hip/hip_runtime.h>
#include <cstdint>
#include <cstddef>

// Problem constants (from the reference).
#define BATCH      4096
#define N_CTX      8
#define N_TGT      16
#define NUM_WORDS  8000
#define VEC_DIM    128
#define WAVES_PER_BLOCK 8   // 256 threads = 8 wave32's, one batch row per wave

typedef __attribute__((ext_vector_type(2))) float v2f;
typedef __attribute__((ext_vector_type(4))) float v4f;
typedef __attribute__((ext_vector_type(8))) float v8f;

#if __has_builtin(__builtin_amdgcn_wmma_f32_16x16x4_f32)
#define HAVE_WMMA_F32_16X16X4 1
#else
#define HAVE_WMMA_F32_16X16X4 0
#endif

// ---------------------------------------------------------------------------
// Kernel 0: LDS-tiled transpose of O (128 x 8000) -> Ot (8000 x 128) in d_ws.
// Makes the per-(b,t) column gathers of O contiguous 512B row gathers.
// One-time cost: ~8 MB of traffic, trivially amortized.
// ---------------------------------------------------------------------------
__global__ __launch_bounds__(256) void o_transpose(const float* __restrict__ O,
                                                   float* __restrict__ Ot) {
  __shared__ float tile[32][33];  // +1 pad: conflict-free transpose
  const int w0 = blockIdx.x * 32;  // word tile
  const int v0 = blockIdx.y * 32;  // vec-dim tile
#pragma unroll
  for (int i = 0; i < 32; i += 8)
    tile[threadIdx.y + i][threadIdx.x] =
        O[(size_t)(v0 + threadIdx.y + i) * NUM_WORDS + (w0 + threadIdx.x)];
  __syncthreads();
#pragma unroll
  for (int i = 0; i < 32; i += 8)
    Ot[(size_t)(w0 + threadIdx.y + i) * VEC_DIM + (v0 + threadIdx.x)] =
        tile[threadIdx.x][threadIdx.y + i];
}

// ---------------------------------------------------------------------------
// Kernel 1: fused gather + ctx-sum + per-batch (1x128)·(128x16) via WMMA.
// One wave per batch row. Stage 1: each lane accumulates a float4 slice of
// x = sum_c (D[doc,ctx_c,:] + W[ctx_c,:]) with fully coalesced 512B row reads.
// Stage 2: A = 16 gathered o-vectors (lane t holds row t), B = x broadcast to
// all 16 columns (LDS scalar broadcast), 32x V_WMMA_F32_16X16X4_F32 over K=128.
// ---------------------------------------------------------------------------
template <bool USE_OT>
__global__ __launch_bounds__(256) void dm_fused(
    const int* __restrict__ ctx_ids, const int* __restrict__ doc_ids,
    const int* __restrict__ tgt_ids, const float* __restrict__ D,
    const float* __restrict__ W, const float* __restrict__ O,
    const float* __restrict__ Ot, float* __restrict__ out) {
  __shared__ __align__(16) float xs[WAVES_PER_BLOCK * VEC_DIM];  // x per wave

  const int wave = threadIdx.x >> 5;
  const int lane = threadIdx.x & 31;
  const int b = blockIdx.x * WAVES_PER_BLOCK + wave;  // batch row of this wave

  const size_t dbase = (size_t)doc_ids[b] * ((size_t)NUM_WORDS * VEC_DIM);

  // ---- Stage 1: x[b, lane*4 .. lane*4+3] ----
  v4f x4 = {0.f, 0.f, 0.f, 0.f};
#pragma unroll
  for (int c = 0; c < N_CTX; ++c) {
    const int w = ctx_ids[b * N_CTX + c];
    const v4f dv = *(const v4f*)(D + dbase + (size_t)w * VEC_DIM + lane * 4);
    const v4f wv = *(const v4f*)(W + (size_t)w * VEC_DIM + lane * 4);
    x4 += dv + wv;
  }
  *(v4f*)&xs[wave * VEC_DIM + lane * 4] = x4;
  // Wave-local LDS producer->consumer: LDS is in-order per wave; the memory
  // clobber also pins compiler ordering of the ds_store before the ds_loads.
  asm volatile("s_wait_dscnt 0x0" ::: "memory");

  // ---- Stage 2: out[b, 0..15] = Og(16x128) @ x(128) ----
  const int t   = lane & 15;        // target row this lane carries in A
  const int hi  = (lane >> 4) & 1;  // K sub-pair select (ISA A-matrix layout)
  const int tid = tgt_ids[b * N_TGT + t];
  const float* xrow = &xs[wave * VEC_DIM];

#if HAVE_WMMA_F32_16X16X4
  v8f acc = {0.f, 0.f, 0.f, 0.f, 0.f, 0.f, 0.f, 0.f};
#pragma unroll
  for (int k0 = 0; k0 < VEC_DIM; k0 += 4) {
    // A 16x4 tile: lane<16 holds {K=k0,k0+1} of row t; lane>=16 holds {k0+2,k0+3}.
    v2f a;
    if (USE_OT) {
      a = *(const v2f*)(Ot + (size_t)tid * VEC_DIM + k0 + 2 * hi);
    } else {
      a.x = O[(size_t)(k0 + 2 * hi) * NUM_WORDS + tid];
      a.y = O[(size_t)(k0 + 2 * hi + 1) * NUM_WORDS + tid];
    }
    // B 4x16 tile with every column = x: mirrored K layout, LDS broadcast read.
    const v2f bv = *(const v2f*)(xrow + k0 + 2 * hi);
    acc = __builtin_amdgcn_wmma_f32_16x16x4_f32(false, a, false, bv,
                                                (short)0, acc, false, false);
  }
  // D layout: lane n (0-15) VGPR r = D[M=r, N=n]; lanes 16-31: M=r+8.
  // All 16 columns identical -> lane 0 holds out[b,0..7], lane 16 out[b,8..15].
  if ((lane & 15) == 0) {
    float* op = out + (size_t)b * N_TGT + 8 * hi;
    v4f lo4 = {acc[0], acc[1], acc[2], acc[3]};
    v4f hi4 = {acc[4], acc[5], acc[6], acc[7]};
    *(v4f*)op = lo4;
    *(v4f*)(op + 4) = hi4;
  }
#else
  // Fallback: per-lane half-dot + one xor-16 shuffle combine.
  float partial = 0.f;
  const int vbase = hi * 64;
  if (USE_OT) {
    const float* orow = Ot + (size_t)tid * VEC_DIM + vbase;
#pragma unroll
    for (int j = 0; j < 64; ++j) partial += xrow[vbase + j] * orow[j];
  } else {
#pragma unroll
    for (int j = 0; j < 64; ++j)
      partial += xrow[vbase + j] * O[(size_t)(vbase + j) * NUM_WORDS + tid];
  }
  const float total = partial + __shfl_xor(partial, 16, 32);
  if (lane < 16) out[(size_t)b * N_TGT + lane] = total;
#endif
}

// ---------------------------------------------------------------------------
extern "C" void kernel_launch(void* const* d_in, const int* in_sizes, int n_in,
                              void* d_out, int out_size, void* d_ws,
                              size_t ws_size, hipStream_t stream) {
  const int*   ctx = (const int*)d_in[0];    // context_ids (4096, 8)
  const int*   doc = (const int*)d_in[1];    // doc_ids     (4096,)
  const int*   tgt = (const int*)d_in[2];    // target_noise_ids (4096, 16)
  const float* D   = (const float*)d_in[3];  // (500, 8000, 128)
  const float* W   = (const float*)d_in[4];  // (8000, 128)
  const float* O   = (const float*)d_in[5];  // (128, 8000)
  float* out = (float*)d_out;                // (4096, 16) f32
  (void)in_sizes; (void)n_in; (void)out_size;

  const size_t ot_bytes = (size_t)NUM_WORDS * VEC_DIM * sizeof(float);
  if (ws_size >= ot_bytes) {
    float* Ot = (float*)d_ws;
    o_transpose<<<dim3(NUM_WORDS / 32, VEC_DIM / 32), dim3(32, 8), 0, stream>>>(O, Ot);
    dm_fused<true><<<dim3(BATCH / WAVES_PER_BLOCK), dim3(256), 0, stream>>>(
        ctx, doc, tgt, D, W, O, Ot, out);
  } else {
    // Not enough scratch for the O transpose: gather O columns directly
    // (strided, but O is L2-resident at 4 MB).
    dm_fused<false><<<dim3(BATCH / WAVES_PER_BLOCK), dim3(256), 0, stream>>>(
        ctx, doc, tgt, D, W, O, nullptr, out);
  }
}